// DKNN_8856222564939
// MI455X (gfx1250) — compile-verified
//
#include <hip/hip_runtime.h>
#include <math.h>

typedef __attribute__((ext_vector_type(2))) float v2f;
typedef __attribute__((ext_vector_type(8))) float v8f;

#define BQ   128   // queries
#define NN   1024  // neighbors
#define DD   128   // feature dim
#define KTOP 10    // NeuralSort rows summed

// ---------------------------------------------------------------------------
// Kernel 0: row squared-norms for queries and neighbors (tiny, latency-bound).
// ---------------------------------------------------------------------------
__global__ __launch_bounds__(128)
void dknn_norms(const float* __restrict__ q, const float* __restrict__ x,
                float* __restrict__ qn, float* __restrict__ xn) {
    int r = blockIdx.x * blockDim.x + threadIdx.x;
    if (r < BQ) {
        const float* p = q + (size_t)r * DD;
        float s = 0.f;
        #pragma unroll 4
        for (int k = 0; k < DD; ++k) s = fmaf(p[k], p[k], s);
        qn[r] = s;
    } else if (r < BQ + NN) {
        const float* p = x + (size_t)(r - BQ) * DD;
        float s = 0.f;
        #pragma unroll 4
        for (int k = 0; k < DD; ++k) s = fmaf(p[k], p[k], s);
        xn[r - BQ] = s;
    }
}

// ---------------------------------------------------------------------------
// Kernel 1: scores[b][n] = 2*(Q.X^T) - |q|^2 - |x|^2 via fp32 WMMA 16x16x4.
// One wave32 per 16x16 output tile (EXEC all-ones, as WMMA requires).
//   A (16x4 f32): lane holds row M=lane%16, VGPR0/1 = K {0,1} (lanes 0-15)
//                 or K {2,3} (lanes 16-31)  -> contiguous float2 at k+2*half.
//   B (4x16 f32): mirrored, lane holds column N=lane%16 (neighbors row-major
//                 means column n of B is neighbor n's features).
//   C/D (16x16):  c[v] = S[M = v + 8*half][N = lane%16].
// ---------------------------------------------------------------------------
__global__ __launch_bounds__(32)
void dknn_scores(const float* __restrict__ q, const float* __restrict__ x,
                 const float* __restrict__ qn, const float* __restrict__ xn,
                 float* __restrict__ scores) {
    const int ntile = blockIdx.x * 16;
    const int btile = blockIdx.y * 16;
    const int lane  = threadIdx.x;       // 0..31
    const int m     = lane & 15;
    const int half  = lane >> 4;

    const float* qrow = q + (size_t)(btile + m) * DD + 2 * half;
    const float* xrow = x + (size_t)(ntile + m) * DD + 2 * half;

    v8f acc = {};
    #pragma unroll 8
    for (int k = 0; k < DD; k += 4) {
        v2f a = *(const v2f*)(qrow + k);
        v2f b = *(const v2f*)(xrow + k);
        acc = __builtin_amdgcn_wmma_f32_16x16x4_f32(
            /*neg_a=*/false, a, /*neg_b=*/false, b,
            /*c_mod=*/(short)0, acc, /*reuse_a=*/false, /*reuse_b=*/false);
    }

    const float xnc = xn[ntile + m];
    #pragma unroll
    for (int v = 0; v < 8; ++v) {
        const int row = btile + v + 8 * half;
        scores[(size_t)row * NN + (ntile + m)] = 2.0f * acc[v] - qn[row] - xnc;
    }
}

// ---------------------------------------------------------------------------
// Kernel 2: per query row b —
//   A_sum[j] = sum_i |s[j]-s[i]|   (LDS broadcast reads, O(N^2) VALU)
//   out[j]   = sum_{r<K} softmax_j( scaling[r]*s[j] - A_sum[j] )   (tau = 1)
// 256 threads (8 wave32s), each owns 4 strided j's; LDS tree reductions.
// ---------------------------------------------------------------------------
__global__ __launch_bounds__(256)
void dknn_sortsum(const float* __restrict__ scores, float* __restrict__ out) {
    __shared__ float s[NN];
    __shared__ float red[256];

    const int b   = blockIdx.x;
    const int tid = threadIdx.x;
    const float* srow = scores + (size_t)b * NN;

    for (int j = tid; j < NN; j += 256) s[j] = srow[j];
    __syncthreads();

    float my_s[4], my_a[4];
    #pragma unroll
    for (int u = 0; u < 4; ++u) { my_s[u] = s[tid + u * 256]; my_a[u] = 0.f; }

    // A_sum: every thread sweeps all i; s[i] is a broadcast LDS read.
    for (int i = 0; i < NN; ++i) {
        const float si = s[i];
        #pragma unroll
        for (int u = 0; u < 4; ++u) my_a[u] += fabsf(my_s[u] - si);
    }

    float acc_out[4] = {0.f, 0.f, 0.f, 0.f};

    for (int r = 0; r < KTOP; ++r) {
        const float scal = (float)(NN - 1 - 2 * r);   // 1023 - 2r
        float lg[4];
        float lmax = -INFINITY;
        #pragma unroll
        for (int u = 0; u < 4; ++u) {
            lg[u] = scal * my_s[u] - my_a[u];          // TAU == 1
            lmax  = fmaxf(lmax, lg[u]);
        }
        // block max
        red[tid] = lmax;
        __syncthreads();
        for (int off = 128; off > 0; off >>= 1) {
            if (tid < off) red[tid] = fmaxf(red[tid], red[tid + off]);
            __syncthreads();
        }
        const float gmax = red[0];
        __syncthreads();

        float e[4];
        float lsum = 0.f;
        #pragma unroll
        for (int u = 0; u < 4; ++u) { e[u] = __expf(lg[u] - gmax); lsum += e[u]; }
        // block sum
        red[tid] = lsum;
        __syncthreads();
        for (int off = 128; off > 0; off >>= 1) {
            if (tid < off) red[tid] += red[tid + off];
            __syncthreads();
        }
        const float inv = 1.0f / red[0];
        __syncthreads();

        #pragma unroll
        for (int u = 0; u < 4; ++u) acc_out[u] += e[u] * inv;
    }

    #pragma unroll
    for (int u = 0; u < 4; ++u) out[(size_t)b * NN + tid + u * 256] = acc_out[u];
}

// ---------------------------------------------------------------------------
extern "C" void kernel_launch(void* const* d_in, const int* in_sizes, int n_in,
                              void* d_out, int out_size, void* d_ws, size_t ws_size,
                              hipStream_t stream) {
    const float* q = (const float*)d_in[0];   // [128,128]
    const float* x = (const float*)d_in[1];   // [1024,128]
    float* outp    = (float*)d_out;           // [128,1024]

    // workspace: scores[128*1024] | qnorm[128] | xnorm[1024]
    float* scores = (float*)d_ws;
    float* qn     = scores + (size_t)BQ * NN;
    float* xn     = qn + BQ;

    dknn_norms  <<<dim3((BQ + NN + 127) / 128), dim3(128), 0, stream>>>(q, x, qn, xn);
    dknn_scores <<<dim3(NN / 16, BQ / 16),      dim3(32),  0, stream>>>(q, x, qn, xn, scores);
    dknn_sortsum<<<dim3(BQ),                    dim3(256), 0, stream>>>(scores, outp);
}